// CDBase_24455543783576
// MI455X (gfx1250) — compile-verified
//
#include <hip/hip_runtime.h>

// ---------------------------------------------------------------------------
// Color-difference (Lab, lightness weight 0) for (B,3,512,512) fp32 images.
// Memory-bound streaming kernel: NT 128-bit loads, raw v_log_f32/v_exp_f32 for
// the pow(x,2.4)/cbrt transcendentals, WMMA-based wave32 reductions (gfx1250).
// Roofline: 201 MB streamed once / 23.3 TB/s ~= 8.6 us; ~200M trans ops at
// quarter-rate is the same order, so HW log2/exp2 (not libm) keeps it
// memory-bound.
// ---------------------------------------------------------------------------

typedef float f32x4 __attribute__((ext_vector_type(4)));
typedef float v2f   __attribute__((ext_vector_type(2)));
typedef float v8f   __attribute__((ext_vector_type(8)));

#define HW    262144          // 512*512 pixels per plane
#define V4    65536           // HW / 4 (float4 vectors per plane)
#define BPB   64              // blocks per batch (kernel 1)
#define NTHR  256             // threads per block (8 wave32)

// Raw hardware transcendentals: v_log_f32 is log2, v_exp_f32 is exp2.
__device__ __forceinline__ float hw_log2(float x) { return __builtin_amdgcn_logf(x); }
__device__ __forceinline__ float hw_exp2(float x) { return __builtin_amdgcn_exp2f(x); }

// sRGB -> linear: x<=0.04045 ? x/12.92 : ((x+0.055)/1.055)^2.4
__device__ __forceinline__ float srgb_lin(float x) {
  float p = hw_exp2(2.4f * hw_log2((x + 0.055f) * (1.0f / 1.055f)));
  float l = x * (1.0f / 12.92f);
  return (x <= 0.04045f) ? l : p;
}

// Lab f(): t>delta^3 ? cbrt(t) : t/(3*delta^2) + 4/29
__device__ __forceinline__ float lab_f(float t) {
  float c = hw_exp2(hw_log2(t) * (1.0f / 3.0f));   // t==0 -> exp2(-inf)=0, no NaN
  float l = fmaf(t, 7.787037037f, 0.1379310345f);  // 841/108, 4/29
  return (t > 0.008856451679f) ? c : l;            // (6/29)^3
}

__device__ __forceinline__ float pixel_cd(float r1, float g1, float b1,
                                          float r2, float g2, float b2) {
  float lr1 = srgb_lin(r1), lg1 = srgb_lin(g1), lb1 = srgb_lin(b1);
  float lr2 = srgb_lin(r2), lg2 = srgb_lin(g2), lb2 = srgb_lin(b2);
  // RGB->XYZ with the 1/white divide folded into the matrix rows (compile-time)
  const float m00 = 0.4124564f / 0.95047f, m01 = 0.3575761f / 0.95047f, m02 = 0.1804375f / 0.95047f;
  const float m10 = 0.2126729f,            m11 = 0.7151522f,            m12 = 0.0721750f;
  const float m20 = 0.0193339f / 1.08883f, m21 = 0.1191920f / 1.08883f, m22 = 0.9503041f / 1.08883f;
  float tx1 = fmaf(m00, lr1, fmaf(m01, lg1, m02 * lb1));
  float ty1 = fmaf(m10, lr1, fmaf(m11, lg1, m12 * lb1));
  float tz1 = fmaf(m20, lr1, fmaf(m21, lg1, m22 * lb1));
  float tx2 = fmaf(m00, lr2, fmaf(m01, lg2, m02 * lb2));
  float ty2 = fmaf(m10, lr2, fmaf(m11, lg2, m12 * lb2));
  float tz2 = fmaf(m20, lr2, fmaf(m21, lg2, m22 * lb2));
  float fx1 = lab_f(tx1), fy1 = lab_f(ty1), fz1 = lab_f(tz1);
  float fx2 = lab_f(tx2), fy2 = lab_f(ty2), fz2 = lab_f(tz2);
  // L (weight 0) drops out; a = 500*(fx-fy), b = 200*(fy-fz)
  float da = 500.0f * ((fx1 - fy1) - (fx2 - fy2));
  float db = 200.0f * ((fy1 - fz1) - (fy2 - fz2));
  return sqrtf(fmaf(da, da, db * db));
}

// ---------------------------------------------------------------------------
// Wave32 sum via V_WMMA_F32_16X16X4_F32.
// A(16x4) gets each lane's value in both of its 2 VGPR slots (bijective lane
// mapping -> every A element is some lane's value, each value appears twice).
// B(4x16) = all ones (layout-invariant). D[m][n] = sum_k A[m][k], identical
// for every column n. Per-lane sum of the 8 D VGPRs gives the half-row-set
// total; shfl_xor(16) combines halves; result = 2 * wave_sum, so scale 0.5.
// Must be called with full EXEC (no divergence) per WMMA restriction.
// ---------------------------------------------------------------------------
__device__ __forceinline__ float wave_sum(float s) {
#if defined(__gfx1250__) && __has_builtin(__builtin_amdgcn_wmma_f32_16x16x4_f32)
  v2f a;    a[0] = s;    a[1] = s;
  v2f ones; ones[0] = 1.0f; ones[1] = 1.0f;
  v8f c = {};
  v8f d = __builtin_amdgcn_wmma_f32_16x16x4_f32(
      /*neg_a=*/false, a, /*neg_b=*/false, ones,
      /*c_mod=*/(short)0, c, /*reuse_a=*/false, /*reuse_b=*/false);
  float r = ((d[0] + d[1]) + (d[2] + d[3])) + ((d[4] + d[5]) + (d[6] + d[7]));
  r += __shfl_xor(r, 16, 32);
  return 0.5f * r;
#else
  for (int off = 16; off > 0; off >>= 1) s += __shfl_xor(s, off, 32);
  return s;
#endif
}

// ---------------------------------------------------------------------------
// Kernel 1: per-block partial sums of cd_map. grid = (BPB, nbatch), 256 thr.
// ---------------------------------------------------------------------------
__global__ void __launch_bounds__(NTHR)
cd_partial_kernel(const float* __restrict__ img1,
                  const float* __restrict__ img2,
                  float* __restrict__ partials) {
  const int b = blockIdx.y;
  const f32x4* p1r = (const f32x4*)img1 + (size_t)(b * 3 + 0) * V4;
  const f32x4* p1g = (const f32x4*)img1 + (size_t)(b * 3 + 1) * V4;
  const f32x4* p1b = (const f32x4*)img1 + (size_t)(b * 3 + 2) * V4;
  const f32x4* p2r = (const f32x4*)img2 + (size_t)(b * 3 + 0) * V4;
  const f32x4* p2g = (const f32x4*)img2 + (size_t)(b * 3 + 1) * V4;
  const f32x4* p2b = (const f32x4*)img2 + (size_t)(b * 3 + 2) * V4;

  float s = 0.0f;
  const int stride = BPB * NTHR;                       // 16384 -> 4 iterations
  for (int i = blockIdx.x * NTHR + threadIdx.x; i < V4; i += stride) {
    // streamed exactly once: non-temporal 128-bit loads (global_load_b128, NT)
    f32x4 r1 = __builtin_nontemporal_load(p1r + i);
    f32x4 g1 = __builtin_nontemporal_load(p1g + i);
    f32x4 b1 = __builtin_nontemporal_load(p1b + i);
    f32x4 r2 = __builtin_nontemporal_load(p2r + i);
    f32x4 g2 = __builtin_nontemporal_load(p2g + i);
    f32x4 b2 = __builtin_nontemporal_load(p2b + i);
#pragma unroll
    for (int k = 0; k < 4; ++k)
      s += pixel_cd(r1[k], g1[k], b1[k], r2[k], g2[k], b2[k]);
  }

  // uniform control flow up to here -> EXEC all-ones for the WMMA reduce
  float ws = wave_sum(s);

  __shared__ float lds[NTHR / 32];
  const int wid  = threadIdx.x >> 5;
  const int lane = threadIdx.x & 31;
  if (lane == 0) lds[wid] = ws;
  __syncthreads();
  if (threadIdx.x == 0) {
    float t = 0.0f;
#pragma unroll
    for (int w = 0; w < NTHR / 32; ++w) t += lds[w];
    partials[b * BPB + blockIdx.x] = t;
  }
}

// ---------------------------------------------------------------------------
// Kernel 2: reduce BPB partials per batch -> mean. grid = nbatch, 64 threads.
// ---------------------------------------------------------------------------
__global__ void __launch_bounds__(BPB)
cd_final_kernel(const float* __restrict__ partials, float* __restrict__ out) {
  const int b = blockIdx.x;
  float v = partials[b * BPB + threadIdx.x];
  float ws = wave_sum(v);                 // full EXEC, uniform flow
  __shared__ float lds[BPB / 32];
  const int wid  = threadIdx.x >> 5;
  const int lane = threadIdx.x & 31;
  if (lane == 0) lds[wid] = ws;
  __syncthreads();
  if (threadIdx.x == 0)
    out[b] = (lds[0] + lds[1]) * (1.0f / (float)HW);  // exact pow2 scale
}

// ---------------------------------------------------------------------------
extern "C" void kernel_launch(void* const* d_in, const int* in_sizes, int n_in,
                              void* d_out, int out_size, void* d_ws, size_t ws_size,
                              hipStream_t stream) {
  const float* img1 = (const float*)d_in[0];
  const float* img2 = (const float*)d_in[1];
  float* out      = (float*)d_out;
  float* partials = (float*)d_ws;               // nbatch * BPB floats (8 KB @ B=32)

  const int nbatch = in_sizes[0] / (3 * HW);    // 32 for the reference shapes

  dim3 grid1(BPB, nbatch);
  cd_partial_kernel<<<grid1, NTHR, 0, stream>>>(img1, img2, partials);
  cd_final_kernel<<<nbatch, BPB, 0, stream>>>(partials, out);
}